// EdgeNetwork_30236569764494
// MI455X (gfx1250) — compile-verified
//
#include <hip/hip_runtime.h>

typedef float v2f __attribute__((ext_vector_type(2)));
typedef float v8f __attribute__((ext_vector_type(8)));

namespace {
constexpr int Bn   = 8;     // batch
constexpr int Nn   = 64;    // nodes
constexpr int Dd   = 50;    // node dim
constexpr int HDim = 50;    // MLP hidden dim
constexpr int KPAD = 52;    // hidden dim padded to multiple of 4 (WMMA K-chunks)
constexpr int KKn  = Nn * KPAD;   // 3328 total contraction length (w,k)
constexpr int IPAD = 64;    // output-i padded to 4 WMMA N-tiles
constexpr int NCHUNK = 4;   // split-K chunks (w in [c*16, c*16+16))
constexpr int TOT_STEPS = KKn / 4;                 // 832
constexpr int STEPS_PER_CHUNK = TOT_STEPS / NCHUNK; // 208

// workspace layout (floats)
constexpr size_t H4SZ   = (size_t)Bn * Nn * Nn * KPAD;   // h4[b][v][w*52+k]
constexpr size_t P2SZ   = (size_t)Bn * KKn * IPAD;       // P, K-pair interleaved
constexpr size_t CBSZ   = 512;                           // cbias[b][50] (rounded)
constexpr size_t PARTSZ = (size_t)NCHUNK * Bn * Nn * 2 * Dd;
}

// ---------------------------------------------------------------------------
// Kernel 0: cbias[b][i] = sum_j B5[i][j] * (sum_w ns[b][w][j])
// ---------------------------------------------------------------------------
__global__ void bias_kernel(const float* __restrict__ ns,
                            const float* __restrict__ b5,
                            float* __restrict__ cbias) {
  __shared__ float lns[Dd];
  const int b = blockIdx.x;
  const int t = threadIdx.x;
  if (t < Dd) {
    float s = 0.f;
    for (int w = 0; w < Nn; ++w) s += ns[((size_t)b * Nn + w) * Dd + t];
    lns[t] = s;
  }
  __syncthreads();
  if (t < Dd) {
    float s = 0.f;
    for (int j = 0; j < Dd; ++j) s += b5[t * Dd + j] * lns[j];
    cbias[b * Dd + t] = s;
  }
}

// ---------------------------------------------------------------------------
// Kernel 1: P[b,w,k,i] = sum_j W5[k][i*50+j] * ns[b][w][j]
// Stored interleaved by K-pairs so a WMMA B-fragment is one float2 per lane:
//   (b,kk,i) -> ((b*KK/2 + kk/2)*IPAD + i)*2 + (kk&1),  kk = w*52 + k
// ---------------------------------------------------------------------------
__global__ void pmat_kernel(const float* __restrict__ ns,
                            const float* __restrict__ W5,
                            float* __restrict__ P2) {
  __shared__ float lns[Dd];
  const int node = blockIdx.x;          // b*Nn + w
  const int b = node >> 6, w = node & 63;
  const int t = threadIdx.x;
  if (t < Dd) lns[t] = ns[(size_t)node * Dd + t];
  __syncthreads();
  for (int p = t; p < KPAD * IPAD; p += blockDim.x) {
    const int k = p >> 6;               // 0..51
    const int i = p & 63;               // 0..63
    float s = 0.f;
    if (k < HDim && i < Dd) {
      const float* wrow = W5 + (size_t)k * (Dd * Dd) + (size_t)i * Dd;
      for (int j = 0; j < Dd; ++j) s += wrow[j] * lns[j];
    }
    const int kk = w * KPAD + k;
    P2[(((size_t)b * (KKn / 2) + (kk >> 1)) * IPAD + i) * 2 + (kk & 1)] = s;
  }
}

// ---------------------------------------------------------------------------
// Kernel 2: per-edge 4-layer MLP -> h4[edge*52 + k]  (k>=50 zero-padded)
// First layer exploits the one-hot input: h1 = relu(W1[adj] + dist*W1[3] + b1)
// ---------------------------------------------------------------------------
__global__ void mlp_kernel(const int* __restrict__ adj,
                           const float* __restrict__ dist,
                           const float* __restrict__ W1, const float* __restrict__ b1,
                           const float* __restrict__ W2, const float* __restrict__ b2,
                           const float* __restrict__ W3, const float* __restrict__ b3,
                           const float* __restrict__ W4, const float* __restrict__ b4,
                           float* __restrict__ h4out) {
  __shared__ float ha[HDim], hb[HDim];
  const int e = blockIdx.x;             // (b*64+v)*64+w
  const int t = threadIdx.x;
  const int a = adj[e];
  const float x3 = dist[e];
  if (t < HDim) {
    float v = W1[a * HDim + t] + x3 * W1[3 * HDim + t] + b1[t];
    ha[t] = v > 0.f ? v : 0.f;
  }
  __syncthreads();
  if (t < HDim) {
    float s = b2[t];
    for (int j = 0; j < HDim; ++j) s += ha[j] * W2[j * HDim + t];
    hb[t] = s > 0.f ? s : 0.f;
  }
  __syncthreads();
  if (t < HDim) {
    float s = b3[t];
    for (int j = 0; j < HDim; ++j) s += hb[j] * W3[j * HDim + t];
    ha[t] = s > 0.f ? s : 0.f;
  }
  __syncthreads();
  if (t < HDim) {
    float s = b4[t];
    for (int j = 0; j < HDim; ++j) s += ha[j] * W4[j * HDim + t];
    hb[t] = s > 0.f ? s : 0.f;
  }
  __syncthreads();
  if (t < KPAD) h4out[(size_t)e * KPAD + t] = (t < HDim) ? hb[t] : 0.f;
}

// ---------------------------------------------------------------------------
// Kernel 3: WMMA contraction.
//   m_in [b,v,i] += sum_kk h4[b][v][kk]        * P[b][kk][i]
//   m_out[b,v,i] += sum_kk h4[b][w][v*52 + k]  * P[b][kk][i]   (same B operand)
// One wave per (chunk, v-tile, b). f32 WMMA 16x16x4, 8 WMMAs per K-step.
// Partials go to ws; reduce_kernel sums chunks + biases.
// ---------------------------------------------------------------------------
__global__ void __launch_bounds__(32)
msg_kernel(const float* __restrict__ h4, const float* __restrict__ P2,
           float* __restrict__ part) {
  const int c     = blockIdx.x;         // split-K chunk
  const int vt    = blockIdx.y;         // v tile (16 rows)
  const int b     = blockIdx.z;
  const int lane  = threadIdx.x;
  const int m     = lane & 15;
  const int half  = lane >> 4;          // 0/1 -> K-subpair / row-half
  const int vbase = vt * 16;

  v8f accI[4], accO[4];
#pragma unroll
  for (int t = 0; t < 4; ++t) { accI[t] = {}; accO[t] = {}; }

  const float* aInRow = h4 + (size_t)(b * Nn + vbase + m) * KKn;
  const float* h4b    = h4 + (size_t)b * Nn * KKn;
  const float* p2b    = P2 + (size_t)b * (KKn / 2) * IPAD * 2;

  const int s0 = c * STEPS_PER_CHUNK;
  for (int s = s0; s < s0 + STEPS_PER_CHUNK; ++s) {
    const int kk0   = s * 4;
    const int w     = s / 13;           // kk0 / 52  (52 % 4 == 0)
    const int kbase = (s % 13) * 4;

    // A (16x4 f32): lanes 0-15 {K=0,K=1}, lanes 16-31 {K=2,K=3}
    v2f aI = *(const v2f*)(aInRow + kk0 + 2 * half);
    v2f aO = *(const v2f*)(h4b + (size_t)w * KKn + (vbase + m) * KPAD + kbase + 2 * half);

    const float* pbase = p2b + ((size_t)(kk0 >> 1) + half) * (IPAD * 2) + m * 2;
#pragma unroll
    for (int t = 0; t < 4; ++t) {
      // B (4x16 f32): lane holds {B[K=2*half][n], B[K=2*half+1][n]}, n = t*16+m
      v2f bf = *(const v2f*)(pbase + t * 32);
      accI[t] = __builtin_amdgcn_wmma_f32_16x16x4_f32(
          false, aI, false, bf, (short)0, accI[t], false, false);
      accO[t] = __builtin_amdgcn_wmma_f32_16x16x4_f32(
          false, aO, false, bf, (short)0, accO[t], false, false);
    }
  }

  // C/D layout: lane n = lane&15, rows = half*8 + r
  float* pc = part + (size_t)c * (Bn * Nn * 2 * Dd);
#pragma unroll
  for (int t = 0; t < 4; ++t) {
    const int col = t * 16 + m;
    if (col < Dd) {
#pragma unroll
      for (int r = 0; r < 8; ++r) {
        const int v = vbase + half * 8 + r;
        const size_t o = (size_t)(b * Nn + v) * (2 * Dd);
        pc[o + col]      = accI[t][r];
        pc[o + Dd + col] = accO[t][r];
      }
    }
  }
}

// ---------------------------------------------------------------------------
// Kernel 4: sum split-K partials + b5-bias + message_bias -> out[b][v][100]
// ---------------------------------------------------------------------------
__global__ void reduce_kernel(const float* __restrict__ part,
                              const float* __restrict__ cbias,
                              const float* __restrict__ mbias,
                              float* __restrict__ out) {
  const int idx = blockIdx.x * blockDim.x + threadIdx.x;
  const int total = Bn * Nn * 2 * Dd;
  if (idx >= total) return;
  const int b      = idx / (Nn * 2 * Dd);
  const int col100 = idx % (2 * Dd);
  const int col    = (col100 < Dd) ? col100 : col100 - Dd;
  float s = cbias[b * Dd + col] + mbias[col100];
#pragma unroll
  for (int c = 0; c < NCHUNK; ++c) s += part[(size_t)c * total + idx];
  out[idx] = s;
}

// ---------------------------------------------------------------------------
extern "C" void kernel_launch(void* const* d_in, const int* in_sizes, int n_in,
                              void* d_out, int out_size, void* d_ws, size_t ws_size,
                              hipStream_t stream) {
  const float* ns    = (const float*)d_in[0];
  const int*   adj   = (const int*)  d_in[1];
  const float* dist  = (const float*)d_in[2];
  const float* W1    = (const float*)d_in[3];
  const float* b1    = (const float*)d_in[4];
  const float* W2    = (const float*)d_in[5];
  const float* b2    = (const float*)d_in[6];
  const float* W3    = (const float*)d_in[7];
  const float* b3    = (const float*)d_in[8];
  const float* W4    = (const float*)d_in[9];
  const float* b4    = (const float*)d_in[10];
  const float* W5    = (const float*)d_in[11];
  const float* b5    = (const float*)d_in[12];
  const float* mbias = (const float*)d_in[13];
  float* out = (float*)d_out;

  float* ws   = (float*)d_ws;
  float* h4   = ws;
  float* P2   = ws + H4SZ;
  float* cb   = ws + H4SZ + P2SZ;
  float* part = ws + H4SZ + P2SZ + CBSZ;

  hipLaunchKernelGGL(bias_kernel, dim3(Bn), dim3(64), 0, stream, ns, b5, cb);
  hipLaunchKernelGGL(pmat_kernel, dim3(Bn * Nn), dim3(256), 0, stream, ns, W5, P2);
  hipLaunchKernelGGL(mlp_kernel, dim3(Bn * Nn * Nn), dim3(64), 0, stream,
                     adj, dist, W1, b1, W2, b2, W3, b3, W4, b4, h4);
  hipLaunchKernelGGL(msg_kernel, dim3(NCHUNK, Nn / 16, Bn), dim3(32), 0, stream,
                     h4, P2, part);
  const int total = Bn * Nn * 2 * Dd;
  hipLaunchKernelGGL(reduce_kernel, dim3((total + 255) / 256), dim3(256), 0, stream,
                     part, cb, mbias, out);
}